// WRNN_77687368450200
// MI455X (gfx1250) — compile-verified
//
#include <hip/hip_runtime.h>

#define N_NODES 100000
#define N_EDGES 640000
#define D       128
#define LDS_STRIDE 132   // 128 + 4 floats pad -> conflict-free b64 B-fragment reads

typedef float v2f __attribute__((ext_vector_type(2)));
typedef float v8f __attribute__((ext_vector_type(8)));

__global__ __launch_bounds__(256) void WRNN_zero_kernel(float* __restrict__ out, int n4) {
    int t = blockIdx.x * blockDim.x + threadIdx.x;
    if (t < n4) ((float4*)out)[t] = make_float4(0.f, 0.f, 0.f, 0.f);
}

// h = x @ W^T using V_WMMA_F32_16X16X4_F32.
// One wave computes 16 output rows x 128 cols (8 tiles of 16x16).
__global__ __launch_bounds__(128) void WRNN_gemm_kernel(const float* __restrict__ x,
                                                        const float* __restrict__ W,
                                                        float* __restrict__ h) {
    __shared__ float lds[D * LDS_STRIDE];

    const int tid = threadIdx.x;

    // Stage W into padded LDS, coalesced: 4096 float4's across 128 threads.
    #pragma unroll
    for (int it = 0; it < 32; ++it) {
        int i4   = tid + 128 * it;          // float4 index into W
        int row  = i4 >> 5;                 // 32 float4 per 128-float row
        int col4 = (i4 & 31) << 2;
        float4 w = ((const float4*)W)[i4];
        *(float4*)&lds[row * LDS_STRIDE + col4] = w;
    }
    __syncthreads();

    const int lane      = tid & 31;
    const int waveId    = blockIdx.x * 4 + (tid >> 5);
    const int rowBase   = waveId * 16;      // 100000/16 = 6250 waves exactly
    if (rowBase >= N_NODES) return;

    const int m  = lane & 15;               // M index (A) / N index (B)
    const int kb = (lane >> 4) << 1;        // lower half lanes: K+0/1, upper: K+2/3

    // Preload A fragments for all 32 K-steps: lane holds x[rowBase+m][4s+kb .. +1]
    const float* xr = x + (size_t)(rowBase + m) * D + kb;
    v2f a_frag[32];
    #pragma unroll
    for (int s = 0; s < 32; ++s)
        a_frag[s] = *(const v2f*)(xr + 4 * s);

    v8f acc[8] = {};

    #pragma unroll
    for (int s = 0; s < 32; ++s) {
        const int k = 4 * s + kb;
        #pragma unroll
        for (int nt = 0; nt < 8; ++nt) {
            // B[k][n] = W[n][k]; lane reads W[nt*16+m][k], W[nt*16+m][k+1]
            v2f b = *(const v2f*)&lds[(nt * 16 + m) * LDS_STRIDE + k];
            acc[nt] = __builtin_amdgcn_wmma_f32_16x16x4_f32(
                /*neg_a=*/false, a_frag[s], /*neg_b=*/false, b,
                /*c_mod=*/(short)0, acc[nt], /*reuse_a=*/false, /*reuse_b=*/false);
        }
    }

    // C/D layout: VGPR r -> lanes 0-15: row M=r, lanes 16-31: row M=8+r
    float* hb = h + (size_t)rowBase * D;
    const int rOff = (lane >> 4) * 8;
    #pragma unroll
    for (int nt = 0; nt < 8; ++nt) {
        #pragma unroll
        for (int r = 0; r < 8; ++r) {
            hb[(size_t)(rOff + r) * D + nt * 16 + m] = acc[nt][r];
        }
    }
}

// out[dst[e]] += h[src[e]] : one wave per edge, one float4 per lane.
__global__ __launch_bounds__(256) void WRNN_scatter_kernel(const float* __restrict__ h,
                                                           const long long* __restrict__ ei,
                                                           float* __restrict__ out) {
    unsigned t = blockIdx.x * blockDim.x + threadIdx.x;
    unsigned e = t >> 5;
    unsigned l = t & 31;
    if (e >= N_EDGES) return;

    long long s = ei[e];
    long long d = ei[N_EDGES + e];

    float4 v = ((const float4*)(h + (size_t)s * D))[l];
    float* o = out + (size_t)d * D + (l << 2);
    atomicAdd(o + 0, v.x);
    atomicAdd(o + 1, v.y);
    atomicAdd(o + 2, v.z);
    atomicAdd(o + 3, v.w);
}

extern "C" void kernel_launch(void* const* d_in, const int* in_sizes, int n_in,
                              void* d_out, int out_size, void* d_ws, size_t ws_size,
                              hipStream_t stream) {
    const float*     x  = (const float*)d_in[0];      // [100000, 128] fp32
    const float*     W  = (const float*)d_in[1];      // [128, 128]   fp32
    const long long* ei = (const long long*)d_in[2];  // [2, 640000]  int64
    float* out = (float*)d_out;                       // [100000, 128] fp32
    float* h   = (float*)d_ws;                        // scratch: 51.2 MB for h

    // 1) zero output (harness poisons it)
    const int n4 = (N_NODES * D) / 4;                 // 3.2M float4
    WRNN_zero_kernel<<<(n4 + 255) / 256, 256, 0, stream>>>(out, n4);

    // 2) h = x @ W^T  (6250 waves, 4 waves/block)
    WRNN_gemm_kernel<<<(6250 + 3) / 4, 128, 0, stream>>>(x, W, h);

    // 3) scatter-add over edges (1 wave per edge)
    WRNN_scatter_kernel<<<(N_EDGES * 32) / 256, 256, 0, stream>>>(h, ei, out);
}